// DisentangledHead_45518063403589
// MI455X (gfx1250) — compile-verified
//
#include <hip/hip_runtime.h>

typedef __attribute__((ext_vector_type(2))) float v2f;
typedef __attribute__((ext_vector_type(8))) float v8f;

#define MD 64          // MODEL_DIM
#define HH 64          // HEAD_SIZE
#define BB 8
#define TT 2048
#define SCALE 0.125f   // 64^-0.5

// ---------------------------------------------------------------------------
// V_WMMA_F32_16X16X4_F32: D(16x16,f32) = A(16x4,f32) * B(4x16,f32) + C
// 8-arg form: (neg_a, A, neg_b, B, c_mod, C, reuse_a, reuse_b)
// ---------------------------------------------------------------------------
__device__ __forceinline__ v8f wmma4(v2f a, v2f b, v8f c) {
  return __builtin_amdgcn_wmma_f32_16x16x4_f32(false, a, false, b, (short)0, c,
                                               false, false);
}

// A-fragment (16x4) from row-major X[ld]: rows m0..m0+15, cols kk..kk+3.
// Lane L: M = L%16, K = {0,1} + 2*(L/16)  ->  one float2 load.
// (Identical addressing serves B-fragments of X^T, i.e. B[k,n]=X[n,k].)
__device__ __forceinline__ v2f loadA(const float* __restrict__ X, int ld,
                                     int m0, int kk, int lane) {
  const float* p = X + (size_t)(m0 + (lane & 15)) * ld + kk + 2 * (lane >> 4);
  v2f v; v[0] = p[0]; v[1] = p[1];
  return v;
}

// B-fragment (4x16) from row-major W[ld]: rows kk..kk+3 (= contraction),
// cols n0..n0+15.  Lane L: N = L%16, K = {0,1} + 2*(L/16) -> 2 strided loads.
__device__ __forceinline__ v2f loadB(const float* __restrict__ W, int ld,
                                     int kk, int n0, int lane) {
  const float* p = W + (size_t)(kk + 2 * (lane >> 4)) * ld + n0 + (lane & 15);
  v2f v; v[0] = p[0]; v[1] = p[ld];
  return v;
}

__device__ __forceinline__ v8f vzero8() {
  v8f z = {0.f, 0.f, 0.f, 0.f, 0.f, 0.f, 0.f, 0.f};
  return z;
}

// ---------------------------------------------------------------------------
// Kernel 1: projections.  One wave per 16-row tile.
//   waves [0,1024):    content tile -> QC, KC, V
//   waves [1024,1152): position tile -> QP, KP
// ---------------------------------------------------------------------------
__device__ __forceinline__ void projOne(const v2f a[16],
                                        const float* __restrict__ W,
                                        float* __restrict__ O, int m0,
                                        int lane) {
  const int half = lane >> 4, nl = lane & 15;
#pragma unroll
  for (int nt = 0; nt < 4; nt++) {
    v8f acc = vzero8();
#pragma unroll
    for (int s = 0; s < 16; s++)
      acc = wmma4(a[s], loadB(W, HH, 4 * s, nt * 16, lane), acc);
#pragma unroll
    for (int r = 0; r < 8; r++)
      O[(size_t)(m0 + r + 8 * half) * HH + nt * 16 + nl] = acc[r];
  }
}

__global__ __launch_bounds__(256) void proj_kernel(
    const float* __restrict__ content, const float* __restrict__ position,
    const float* __restrict__ w_qc, const float* __restrict__ w_kc,
    const float* __restrict__ w_v, const float* __restrict__ w_qp,
    const float* __restrict__ w_kp, float* __restrict__ ws) {
  const int wave = (int)((blockIdx.x * blockDim.x + threadIdx.x) >> 5);
  const int lane = (int)(threadIdx.x & 31);
  float* QC = ws;
  float* KC = ws + (size_t)BB * TT * HH;
  float* Vo = ws + 2 * (size_t)BB * TT * HH;
  float* QP = ws + 3 * (size_t)BB * TT * HH;
  float* KP = QP + (size_t)TT * HH;

  const int nContent = BB * TT / 16;  // 1024
  const int nPos = TT / 16;           // 128
  if (wave >= nContent + nPos) return;

  if (wave < nContent) {
    const int m0 = wave * 16;
    v2f a[16];
#pragma unroll
    for (int s = 0; s < 16; s++) a[s] = loadA(content, MD, m0, 4 * s, lane);
    projOne(a, w_qc, QC, m0, lane);
    projOne(a, w_kc, KC, m0, lane);
    projOne(a, w_v, Vo, m0, lane);
  } else {
    const int m0 = (wave - nContent) * 16;
    v2f a[16];
#pragma unroll
    for (int s = 0; s < 16; s++) a[s] = loadA(position, MD, m0, 4 * s, lane);
    projOne(a, w_qp, QP, m0, lane);
    projOne(a, w_kp, KP, m0, lane);
  }
}

// ---------------------------------------------------------------------------
// Kernel 2: scores + softmax + write attn.
// Block = 512 threads (16 waves) per (b, 16-query tile).
// Wave w owns key columns [w*128, w*128+128): 8 accumulator tiles.
// Score = (QC.KC^T + QP.KP^T) * SCALE accumulated in one WMMA chain.
// ---------------------------------------------------------------------------
__global__ __launch_bounds__(512) void attn_kernel(float* __restrict__ d_out,
                                                   const float* __restrict__ ws) {
  const float* QC = ws;
  const float* KC = ws + (size_t)BB * TT * HH;
  const float* QP = ws + 3 * (size_t)BB * TT * HH;
  const float* KP = QP + (size_t)TT * HH;
  float* attn = d_out + (size_t)BB * TT * HH;

  const int b = (int)(blockIdx.x >> 7);
  const int m0 = ((int)blockIdx.x & 127) * 16;
  const int w = (int)(threadIdx.x >> 5);
  const int lane = (int)(threadIdx.x & 31);
  const int half = lane >> 4, nl = lane & 15;

  const float* qc = QC + (size_t)b * TT * HH;
  const float* kc = KC + (size_t)b * TT * HH;

  v2f aq[16], ap[16];
#pragma unroll
  for (int s = 0; s < 16; s++) {
    aq[s] = loadA(qc, HH, m0, 4 * s, lane);
    ap[s] = loadA(QP, HH, m0, 4 * s, lane);
  }

  v8f acc[8];
#pragma unroll
  for (int ct = 0; ct < 8; ct++) {
    const int n0 = w * 128 + ct * 16;
    v8f c = vzero8();
#pragma unroll
    for (int s = 0; s < 16; s++)  // content term: B = KC^T
      c = wmma4(aq[s], loadA(kc, HH, n0, 4 * s, lane), c);
#pragma unroll
    for (int s = 0; s < 16; s++)  // position term: B = KP^T
      c = wmma4(ap[s], loadA(KP, HH, n0, 4 * s, lane), c);
#pragma unroll
    for (int r = 0; r < 8; r++) c[r] *= SCALE;
    acc[ct] = c;
  }

  __shared__ float sred[16][16];

  // --- row max: intra-wave (16-lane shuffle tree), then cross-wave via LDS
  float pm[8];
#pragma unroll
  for (int r = 0; r < 8; r++) {
    float m = acc[0][r];
#pragma unroll
    for (int ct = 1; ct < 8; ct++) m = fmaxf(m, acc[ct][r]);
#pragma unroll
    for (int d = 1; d < 16; d <<= 1) m = fmaxf(m, __shfl_xor(m, d, 32));
    pm[r] = m;
  }
  if (nl == 0) {
#pragma unroll
    for (int r = 0; r < 8; r++) sred[w][r + 8 * half] = pm[r];
  }
  __syncthreads();
  float gm[8];
#pragma unroll
  for (int r = 0; r < 8; r++) {
    float m = -3.402823466e38f;
    for (int w2 = 0; w2 < 16; w2++) m = fmaxf(m, sred[w2][r + 8 * half]);
    gm[r] = m;
  }
  __syncthreads();

  // --- exp + row sum
  float ps[8];
#pragma unroll
  for (int r = 0; r < 8; r++) {
    float s = 0.f;
#pragma unroll
    for (int ct = 0; ct < 8; ct++) {
      float e = __expf(acc[ct][r] - gm[r]);
      acc[ct][r] = e;
      s += e;
    }
#pragma unroll
    for (int d = 1; d < 16; d <<= 1) s += __shfl_xor(s, d, 32);
    ps[r] = s;
  }
  if (nl == 0) {
#pragma unroll
    for (int r = 0; r < 8; r++) sred[w][r + 8 * half] = ps[r];
  }
  __syncthreads();
#pragma unroll
  for (int r = 0; r < 8; r++) {
    float s = 0.f;
    for (int w2 = 0; w2 < 16; w2++) s += sred[w2][r + 8 * half];
    gm[r] = 1.f / s;  // reuse as inverse denom
  }

  // --- write normalized attn (coalesced across lanes)
  float* arow = attn + (size_t)b * TT * TT;
#pragma unroll
  for (int ct = 0; ct < 8; ct++) {
    const int n = w * 128 + ct * 16 + nl;
#pragma unroll
    for (int r = 0; r < 8; r++)
      arow[(size_t)(m0 + r + 8 * half) * TT + n] = acc[ct][r] * gm[r];
  }
}

// ---------------------------------------------------------------------------
// Kernel 3: out = attn @ V.  One wave per (b, q-tile, 16-col tile), k = 2048.
// ---------------------------------------------------------------------------
__global__ __launch_bounds__(256) void out_kernel(float* __restrict__ d_out,
                                                  const float* __restrict__ ws) {
  const float* V = ws + 2 * (size_t)BB * TT * HH;
  const float* attn = d_out + (size_t)BB * TT * HH;
  const int wave = (int)((blockIdx.x * blockDim.x + threadIdx.x) >> 5);
  const int lane = (int)(threadIdx.x & 31);
  const int n0 = (wave & 3) * 16;
  const int m0 = ((wave >> 2) & 127) * 16;
  const int b = wave >> 9;
  const float* A = attn + (size_t)b * TT * TT;
  const float* Vb = V + (size_t)b * TT * HH;
  const int half = lane >> 4, nl = lane & 15;

  v8f acc = vzero8();
#pragma unroll 4
  for (int kk = 0; kk < TT; kk += 4)
    acc = wmma4(loadA(A, TT, m0, kk, lane), loadB(Vb, HH, kk, n0, lane), acc);

#pragma unroll
  for (int r = 0; r < 8; r++)
    d_out[((size_t)b * TT + m0 + r + 8 * half) * HH + n0 + nl] = acc[r];
}

// ---------------------------------------------------------------------------
extern "C" void kernel_launch(void* const* d_in, const int* in_sizes, int n_in,
                              void* d_out, int out_size, void* d_ws,
                              size_t ws_size, hipStream_t stream) {
  const float* content = (const float*)d_in[0];   // [8,2048,64]
  const float* position = (const float*)d_in[1];  // [2048,64]
  const float* w_qc = (const float*)d_in[2];
  const float* w_kc = (const float*)d_in[3];
  const float* w_v = (const float*)d_in[4];
  const float* w_qp = (const float*)d_in[5];
  const float* w_kp = (const float*)d_in[6];
  float* out = (float*)d_out;  // out[8,2048,64] then attn[8,2048,2048]
  float* ws = (float*)d_ws;    // QC,KC,V [B*T*64] + QP,KP [T*64] = 13.6 MB

  // 1152 wave-tasks, 8 waves/block -> 144 blocks
  proj_kernel<<<144, 256, 0, stream>>>(content, position, w_qc, w_kc, w_v,
                                       w_qp, w_kp, ws);
  // (b, qtile) = 8*128 = 1024 blocks of 16 waves
  attn_kernel<<<1024, 512, 0, stream>>>(out, ws);
  // 8*128*4 = 4096 waves, 8 waves/block -> 512 blocks
  out_kernel<<<512, 256, 0, stream>>>(out, ws);
}